// HungarianMatcher_15874199126550
// MI455X (gfx1250) — compile-verified
//
#include <hip/hip_runtime.h>
#include <hip/hip_bf16.h>

// ---------------------------------------------------------------------------
// HungarianMatcher for MI455X (gfx1250).
//  Stage 0: softmax of class logits -> P  (workspace)
//  Stage 1: cost matrix C[b,q,g] = 5*||p-g||2 - P[b,q,label_g]
//           pairwise ||.||^2 computed with V_WMMA_F32_16X16X4_F32
//           (rank-4 augmented GEMM trick: A=[x,y,|p|^2,1], B=[-2x,-2y,1,|g|^2])
//  Stage 2: exact Jonker-Volgenant Hungarian per batch, 1 workgroup/batch,
//           inner column scan + argmin parallelized over 512 threads in LDS.
// ---------------------------------------------------------------------------

#define BSZ   16
#define NQ    500
#define NG    200
#define NCLS  5
#define QP1   (NQ + 1)   // 501
#define GP1   (NG + 1)   // 201
#define QPAD  512        // padded Q for transposed-cost scratch rows
#define QT    32         // ceil(500/16)
#define GT    13         // ceil(200/16)
#define BIGF  1e9f
#define COST_POINT 5.0f

typedef float v2f __attribute__((ext_vector_type(2)));
typedef float v8f __attribute__((ext_vector_type(8)));

// ---------------- Stage 0: per-query softmax over 5 classes ----------------
__global__ void hm_softmax_kernel(const float* __restrict__ logits,
                                  float* __restrict__ P) {
  int idx = blockIdx.x * blockDim.x + threadIdx.x;   // (b*Q + q)
  if (idx >= BSZ * NQ) return;
  const float* l = logits + idx * NCLS;
  float m = l[0];
  #pragma unroll
  for (int c = 1; c < NCLS; ++c) m = fmaxf(m, l[c]);
  float e[NCLS], s = 0.f;
  #pragma unroll
  for (int c = 0; c < NCLS; ++c) { e[c] = __expf(l[c] - m); s += e[c]; }
  float inv = 1.0f / s;
  float* p = P + idx * NCLS;
  #pragma unroll
  for (int c = 0; c < NCLS; ++c) p[c] = e[c] * inv;
}

// ---------------- Stage 1: cost tiles via WMMA f32 16x16x4 -----------------
// grid = (QT, GT, BSZ), block = 32 (one wave -> one 16x16 tile, one v_wmma)
__global__ void hm_cost_kernel(const float* __restrict__ pred_coords,
                               const float* __restrict__ gt_points,
                               const int*   __restrict__ gt_labels,
                               const float* __restrict__ P,     // softmax probs
                               float* __restrict__ C_out,       // [BSZ,NQ,NG]
                               float* __restrict__ CT,          // [BSZ,NG,QPAD] or null
                               int use_ct) {
  const int tq = blockIdx.x, tg = blockIdx.y, b = blockIdx.z;
  const int lane = threadIdx.x;          // 0..31, wave32
  const int half = lane >> 4;            // 0: K=0,1   1: K=2,3
  const int l16  = lane & 15;

  // ---- A operand: row M = l16 of the query tile, this lane's K pair ----
  int qL = tq * 16 + l16; if (qL > NQ - 1) qL = NQ - 1;
  float qx = pred_coords[(b * NQ + qL) * 2 + 0];
  float qy = pred_coords[(b * NQ + qL) * 2 + 1];
  float nq = qx * qx + qy * qy;
  v2f a;
  a.x = half ? nq   : qx;    // K=0 or K=2
  a.y = half ? 1.0f : qy;    // K=1 or K=3

  // ---- B operand: column N = l16 of the gt tile, this lane's K pair ----
  int gL = tg * 16 + l16; if (gL > NG - 1) gL = NG - 1;
  float gx = gt_points[(b * NG + gL) * 2 + 0];
  float gy = gt_points[(b * NG + gL) * 2 + 1];
  float ng = gx * gx + gy * gy;
  v2f bm;
  bm.x = half ? 1.0f : (-2.0f * gx);
  bm.y = half ? ng   : (-2.0f * gy);

  v8f acc = {};
  // D(m,n) = -2(qx*gx + qy*gy) + |q|^2 + |g|^2 = squared distance
  acc = __builtin_amdgcn_wmma_f32_16x16x4_f32(
      /*neg_a=*/false, a, /*neg_b=*/false, bm,
      /*c_mod=*/(short)0, acc, /*reuse_a=*/false, /*reuse_b=*/false);

  const int gn  = tg * 16 + l16;
  const int lab = gt_labels[b * NG + ((gn < NG) ? gn : (NG - 1))];

  #pragma unroll
  for (int r = 0; r < 8; ++r) {
    int qm = tq * 16 + r + half * 8;       // D layout: lanes>=16 hold M=r+8
    if (qm < NQ && gn < NG) {
      float d2 = acc[r];
      d2 = fmaxf(d2, 0.0f);
      float cost = COST_POINT * __builtin_sqrtf(d2)
                 - P[(b * NQ + qm) * NCLS + lab];
      C_out[((size_t)b * NQ + qm) * NG + gn] = cost;
      if (use_ct)
        CT[((size_t)b * NG + gn) * QPAD + qm] = cost;
    }
  }
}

// ---------------- Stage 2: Jonker-Volgenant Hungarian per batch ------------
// grid = BSZ, block = 512.  cost(i0,j) for row i0 in 1..G, col j in 1..Q.
__global__ __launch_bounds__(512)
void hm_jv_kernel(const float* __restrict__ C_out,   // [BSZ,NQ,NG]
                  const float* __restrict__ CT,      // [BSZ,NG,QPAD] or null
                  int use_ct,
                  float* __restrict__ row_out,       // [BSZ,NG]
                  float* __restrict__ col_out) {     // [BSZ,NG]
  const int b   = blockIdx.x;
  const int tid = threadIdx.x;
  const int nthr = blockDim.x;

  __shared__ float u[GP1];
  __shared__ float v[QP1];
  __shared__ int   p[QP1];
  __shared__ int   way[QP1];
  __shared__ float minv[QP1];
  __shared__ int   used[QP1];
  __shared__ float rval[512];
  __shared__ int   ridx[512];
  __shared__ int   assign[NG];
  __shared__ int   j0s;

  for (int j = tid; j < QP1; j += nthr) { v[j] = 0.f; p[j] = 0; }
  for (int j = tid; j < GP1; j += nthr) u[j] = 0.f;
  for (int j = tid; j < NG;  j += nthr) assign[j] = 0;
  __syncthreads();

  const float* cbat = C_out + (size_t)b * NQ * NG;
  const float* ctb  = use_ct ? (CT + (size_t)b * NG * QPAD) : nullptr;

  for (int i = 1; i <= NG; ++i) {
    if (tid == 0) { p[0] = i; j0s = 0; }
    for (int j = tid; j < QP1; j += nthr) {
      minv[j] = BIGF; used[j] = 0; way[j] = 0;
    }
    __syncthreads();

    // Dijkstra-style search for an augmenting path.
    while (true) {
      __syncthreads();
      const int j0 = j0s;
      const int i0 = p[j0];
      if (i0 == 0) break;
      if (tid == 0) used[j0] = 1;
      __syncthreads();

      // relax all unused columns against row i0
      const float ui0 = u[i0];
      if (tid < QP1 && !used[tid]) {
        float aij;
        if (tid == 0) aij = BIGF;                       // dummy column
        else if (use_ct) aij = ctb[(size_t)(i0 - 1) * QPAD + (tid - 1)];
        else             aij = cbat[(size_t)(tid - 1) * NG + (i0 - 1)];
        float cur = aij - ui0 - v[tid];
        if (cur < minv[tid]) { minv[tid] = cur; way[tid] = j0; }
      }
      // argmin over unused columns
      float m = (tid < QP1 && !used[tid]) ? minv[tid] : BIGF;
      rval[tid] = m; ridx[tid] = tid;
      __syncthreads();
      for (int s = 256; s > 0; s >>= 1) {
        if (tid < s) {
          if (rval[tid + s] < rval[tid]) {
            rval[tid] = rval[tid + s];
            ridx[tid] = ridx[tid + s];
          }
        }
        __syncthreads();
      }
      const float delta = rval[0];
      const int   j1    = ridx[0];

      // dual updates
      if (tid < QP1) {
        if (used[tid]) {
          atomicAdd(&u[p[tid]], delta);  // p injective on used cols; safe anyway
          v[tid] -= delta;
        } else {
          minv[tid] -= delta;
        }
      }
      if (tid == 0) j0s = j1;
      // loop-top __syncthreads orders rval reuse / j0s publish
    }

    // augment along the alternating path (serial pointer chase)
    if (tid == 0) {
      int j0 = j0s;
      while (j0 != 0) {
        int j1 = way[j0];
        p[j0] = p[j1];
        j0 = j1;
      }
    }
    __syncthreads();
  }

  // invert: p[j] = row assigned to col j  ->  assign[row-1] = query j-1
  for (int j = tid; j < NQ; j += nthr) {
    int pr = p[j + 1];
    if (pr > 0) assign[pr - 1] = j;
  }
  __syncthreads();

  // stable argsort via rank-by-count (assignments are distinct)
  for (int i = tid; i < NG; i += nthr) {
    int ai = assign[i];
    int r = 0;
    for (int k = 0; k < NG; ++k) r += (assign[k] < ai) ? 1 : 0;
    row_out[b * NG + r] = (float)ai;   // matched query index
    col_out[b * NG + r] = (float)i;    // matched gt index
  }
}

// ---------------------------------------------------------------------------
extern "C" void kernel_launch(void* const* d_in, const int* in_sizes, int n_in,
                              void* d_out, int out_size, void* d_ws, size_t ws_size,
                              hipStream_t stream) {
  const float* pred_coords = (const float*)d_in[0];  // [16,500,2]
  const float* pred_logits = (const float*)d_in[1];  // [16,500,5]
  const float* gt_points   = (const float*)d_in[2];  // [3200,2]
  const int*   gt_labels   = (const int*)d_in[3];    // [3200]

  float* out     = (float*)d_out;
  float* row_out = out;                      // [16,200]
  float* col_out = out + BSZ * NG;           // [16,200]
  float* C_out   = out + 2 * BSZ * NG;       // [16,500,200]

  // workspace layout: P (softmax probs), then optional transposed cost copy
  const size_t pBytes  = (size_t)BSZ * NQ * NCLS * sizeof(float);
  const size_t ctBytes = (size_t)BSZ * NG * QPAD * sizeof(float);
  float* P  = (float*)d_ws;
  float* CT = (float*)((char*)d_ws + pBytes);
  const int use_ct = (ws_size >= pBytes + ctBytes) ? 1 : 0;

  // Stage 0: softmax
  {
    int n = BSZ * NQ;
    hm_softmax_kernel<<<(n + 255) / 256, 256, 0, stream>>>(pred_logits, P);
  }
  // Stage 1: WMMA cost tiles (one wave per 16x16 tile)
  {
    dim3 grid(QT, GT, BSZ);
    hm_cost_kernel<<<grid, 32, 0, stream>>>(pred_coords, gt_points, gt_labels,
                                            P, C_out, CT, use_ct);
  }
  // Stage 2: Hungarian solve + argsort, one workgroup per batch image
  {
    hm_jv_kernel<<<BSZ, 512, 0, stream>>>(C_out, CT, use_ct, row_out, col_out);
  }
}